// GraphEncoder_74878459838549
// MI455X (gfx1250) — compile-verified
//
#include <hip/hip_runtime.h>

#define D_ 128
#define H_ 128
#define SLOPE 0.2f

typedef __attribute__((ext_vector_type(2))) float v2f;
typedef __attribute__((ext_vector_type(8))) float v8f;

// float atomic max via sign-aware integer atomics (no CAS loop, always compiles)
__device__ __forceinline__ void atomicMaxF(float* a, float v) {
    if (v >= 0.0f) atomicMax((int*)a, __float_as_int(v));
    else           atomicMin((unsigned int*)a, __float_as_uint(v));
}

// ---------------------------------------------------------------- init
__global__ void k_init(float* __restrict__ accum, float* __restrict__ m0, float* __restrict__ m1,
                       float* __restrict__ den0, float* __restrict__ den1, int n, long long total) {
    long long i = (long long)blockIdx.x * blockDim.x + threadIdx.x;
    if (i < total) accum[i] = 0.0f;
    if (i < n) { m0[i] = -3.0e38f; m1[i] = -3.0e38f; den0[i] = 0.0f; den1[i] = 0.0f; }
}

// ---------------------------------------------------------------- v = W @ a  (4 tiny matvecs)
__global__ void k_attnvec(const float* __restrict__ Ws0, const float* __restrict__ a_s0,
                          const float* __restrict__ Wd0, const float* __restrict__ a_d0,
                          const float* __restrict__ Ws1, const float* __restrict__ a_s1,
                          const float* __restrict__ Wd1, const float* __restrict__ a_d1,
                          float* __restrict__ vecs) {
    const float* W; const float* a;
    switch (blockIdx.x) {
        case 0:  W = Ws0; a = a_s0; break;
        case 1:  W = Wd0; a = a_d0; break;
        case 2:  W = Ws1; a = a_s1; break;
        default: W = Wd1; a = a_d1; break;
    }
    int d = threadIdx.x;
    float s = 0.0f;
    for (int h = 0; h < H_; ++h) s += W[d * H_ + h] * a[h];
    vecs[blockIdx.x * H_ + d] = s;
}

// ---------------------------------------------------------------- hs = x @ W  (fp32 WMMA)
// One wave = one 16x16 output tile; K stepped by 4 with V_WMMA_F32_16X16X4_F32.
__global__ void k_proj(const float* __restrict__ x, const float* __restrict__ W0,
                       const float* __restrict__ W1, float* __restrict__ hs0,
                       float* __restrict__ hs1, int n) {
    const float* __restrict__ W   = blockIdx.y ? W1  : W0;
    float* __restrict__       out = blockIdx.y ? hs1 : hs0;
    const int lane = threadIdx.x & 31;
    const int wv   = threadIdx.x >> 5;
    const int half = lane >> 4;    // 0: K pair {0,1}, 1: K pair {2,3}
    const int lid  = lane & 15;
    const int rowBase = blockIdx.x * 16;
    const int colBase = wv * 16;   // 8 waves cover H=128

    int row = rowBase + lid;
    if (row >= n) row = n - 1;     // clamp (stores are guarded)
    const float* xr = x + (size_t)row * D_;

    v8f c = {};
#pragma unroll
    for (int k = 0; k < D_; k += 4) {
        v2f a, b;
        // A 16x4: lanes 0-15 -> K={k,k+1}, lanes 16-31 -> K={k+2,k+3}
        a.x = xr[k + half * 2 + 0];
        a.y = xr[k + half * 2 + 1];
        // B 4x16: lanes 0-15 -> N=lid @ K={k,k+1}, lanes 16-31 -> K={k+2,k+3}
        b.x = W[(size_t)(k + half * 2 + 0) * H_ + colBase + lid];
        b.y = W[(size_t)(k + half * 2 + 1) * H_ + colBase + lid];
        c = __builtin_amdgcn_wmma_f32_16x16x4_f32(false, a, false, b, (short)0, c, false, false);
    }
#pragma unroll
    for (int r = 0; r < 8; ++r) {
        int orow = rowBase + half * 8 + r;     // VGPR r: M=r (lanes<16) / M=r+8
        if (orow < n) out[(size_t)orow * H_ + colBase + lid] = c[r];
    }
}

// ---------------------------------------------------------------- per-node logits a_s, a_d
__global__ void k_logits(const float* __restrict__ x, const float* __restrict__ vecs,
                         float* __restrict__ as0, float* __restrict__ ad0,
                         float* __restrict__ as1, float* __restrict__ ad1, int n) {
    const int lane = threadIdx.x & 31;
    const int wv   = threadIdx.x >> 5;
    int node = blockIdx.x * 8 + wv;
    if (node >= n) return;
    const float4 xv = *(const float4*)(x + (size_t)node * D_ + lane * 4);
    float* outp[4] = {as0, ad0, as1, ad1};
#pragma unroll
    for (int v = 0; v < 4; ++v) {
        const float4 w4 = *(const float4*)(vecs + v * H_ + lane * 4);
        float s = xv.x * w4.x + xv.y * w4.y + xv.z * w4.z + xv.w * w4.w;
#pragma unroll
        for (int o = 16; o > 0; o >>= 1) s += __shfl_xor(s, o, 32);
        if (lane == 0) outp[v][node] = s;
    }
}

// ---------------------------------------------------------------- pass A: leaky_relu + segment max
__global__ void k_edge_max(const int* __restrict__ ei, const float* __restrict__ as,
                           const float* __restrict__ ad, float* __restrict__ m,
                           float* __restrict__ ex, int E, int n) {
    int e = blockIdx.x * blockDim.x + threadIdx.x;
    if (e >= E + n) return;
    int s, d;
    if (e < E) { s = ei[e]; d = ei[E + e]; } else { s = d = e - E; }  // self-loops
    float v = as[s] + ad[d];
    v = (v >= 0.0f) ? v : SLOPE * v;
    ex[e] = v;
    atomicMaxF(m + d, v);
}

// ---------------------------------------------------------------- pass B: exp + segment sum
__global__ void k_edge_exp(const int* __restrict__ ei, const float* __restrict__ m,
                           float* __restrict__ ex, float* __restrict__ den, int E, int n) {
    int e = blockIdx.x * blockDim.x + threadIdx.x;
    if (e >= E + n) return;
    int d = (e < E) ? ei[E + e] : (e - E);
    float p = __expf(ex[e] - m[d]);
    ex[e] = p;
    atomicAdd(den + d, p);
}

// ---------------------------------------------------------------- pass C: alpha-weighted scatter
// One wave32 per edge; lane handles 4 features (float4 gather, 4 atomic adds into L2-resident accum).
__global__ void k_scatter(const int* __restrict__ ei, const float* __restrict__ ex,
                          const float* __restrict__ den, const float* __restrict__ hs,
                          float* __restrict__ accum, int E, int n) {
    const int lane = threadIdx.x & 31;
    const int wv   = threadIdx.x >> 5;
    int e = blockIdx.x * 8 + wv;
    if (e >= E + n) return;
    int s, d;
    if (e < E) { s = ei[e]; d = ei[E + e]; } else { s = d = e - E; }
    float alpha = ex[e] / den[d];
    const float4 h = *(const float4*)(hs + (size_t)s * H_ + lane * 4);
    float* o = accum + (size_t)d * H_ + lane * 4;
    atomicAdd(o + 0, h.x * alpha);
    atomicAdd(o + 1, h.y * alpha);
    atomicAdd(o + 2, h.z * alpha);
    atomicAdd(o + 3, h.w * alpha);
}

// ---------------------------------------------------------------- biases + ELU
__global__ void k_final(const float* __restrict__ accum, const float* __restrict__ b0,
                        const float* __restrict__ b1, float* __restrict__ out, long long total) {
    long long i = (long long)blockIdx.x * blockDim.x + threadIdx.x;
    if (i >= total) return;
    int h = (int)(i & (H_ - 1));
    float v = accum[i] + b0[h] + b1[h];
    out[i] = (v > 0.0f) ? v : (expf(v) - 1.0f);
}

extern "C" void kernel_launch(void* const* d_in, const int* in_sizes, int n_in,
                              void* d_out, int out_size, void* d_ws, size_t ws_size,
                              hipStream_t stream) {
    const float* x    = (const float*)d_in[0];
    const int*   ei0  = (const int*)d_in[1];
    const int*   ei1  = (const int*)d_in[2];
    const float* Ws0  = (const float*)d_in[3];
    const float* Wd0  = (const float*)d_in[4];
    const float* a_s0 = (const float*)d_in[5];
    const float* a_d0 = (const float*)d_in[6];
    const float* b0   = (const float*)d_in[7];
    const float* Ws1  = (const float*)d_in[8];
    const float* Wd1  = (const float*)d_in[9];
    const float* a_s1 = (const float*)d_in[10];
    const float* a_d1 = (const float*)d_in[11];
    const float* b1   = (const float*)d_in[12];
    float* out = (float*)d_out;

    const int n = in_sizes[0] / D_;          // 50000
    const int E = in_sizes[1] / 2;           // 800000
    const long long NH = (long long)n * H_;
    const int EN = E + n;

    // workspace layout (floats)
    float* ws    = (float*)d_ws;
    float* hs0   = ws;
    float* hs1   = hs0 + NH;
    float* accum = hs1 + NH;
    float* as0   = accum + NH;
    float* ad0   = as0 + n;
    float* as1   = ad0 + n;
    float* ad1   = as1 + n;
    float* m0    = ad1 + n;
    float* m1    = m0 + n;
    float* den0  = m1 + n;
    float* den1  = den0 + n;
    float* ex0   = den1 + n;
    float* ex1   = ex0 + EN;
    float* vecs  = ex1 + EN;                 // 4 * 128

    const int tblk = 256;
    const int nhBlocks = (int)((NH + tblk - 1) / tblk);

    k_init<<<nhBlocks, tblk, 0, stream>>>(accum, m0, m1, den0, den1, n, NH);
    k_attnvec<<<4, 128, 0, stream>>>(Ws0, a_s0, Wd0, a_d0, Ws1, a_s1, Wd1, a_d1, vecs);

    dim3 gProj((n + 15) / 16, 2);
    k_proj<<<gProj, 256, 0, stream>>>(x, Ws0, Ws1, hs0, hs1, n);

    k_logits<<<(n + 7) / 8, 256, 0, stream>>>(x, vecs, as0, ad0, as1, ad1, n);

    const int eBlocks = (EN + tblk - 1) / tblk;
    k_edge_max<<<eBlocks, tblk, 0, stream>>>(ei0, as0, ad0, m0, ex0, E, n);
    k_edge_max<<<eBlocks, tblk, 0, stream>>>(ei1, as1, ad1, m1, ex1, E, n);
    k_edge_exp<<<eBlocks, tblk, 0, stream>>>(ei0, m0, ex0, den0, E, n);
    k_edge_exp<<<eBlocks, tblk, 0, stream>>>(ei1, m1, ex1, den1, E, n);

    const int sBlocks = (EN + 7) / 8;
    k_scatter<<<sBlocks, 256, 0, stream>>>(ei0, ex0, den0, hs0, accum, E, n);
    k_scatter<<<sBlocks, 256, 0, stream>>>(ei1, ex1, den1, hs1, accum, E, n);

    k_final<<<nhBlocks, tblk, 0, stream>>>(accum, b0, b1, out, NH);
}